// BinaryCE_wRejectionSMLoss_30399778521687
// MI455X (gfx1250) — compile-verified
//
#include <hip/hip_runtime.h>

// Problem constants (fixed by the reference).
#define BB 256   // batch
#define CC 500   // classes
#define DD 256   // feature dim

typedef __attribute__((ext_vector_type(2))) float v2f;
typedef __attribute__((ext_vector_type(8))) float v8f;

// ---------------------------------------------------------------------------
// Kernel 1: streaming max over D for every (c,b) row of wf [C,B,D].
// One wave32 per contiguous 1KB row; each lane: 2x b128 loads (coalesced
// 512B/instr/wave), 8-way local max, 5-step shfl_xor butterfly.
// Writes maxsim[b*CC + c] into workspace. This kernel carries ~131MB of the
// ~131.5MB total traffic -> it IS the runtime (HBM roofline ~5.6us).
// ---------------------------------------------------------------------------
__global__ __launch_bounds__(256) void rowmax_kernel(const float* __restrict__ wf,
                                                     float* __restrict__ maxsim) {
  const int lane = threadIdx.x & 31;
  const int wave = blockIdx.x * (blockDim.x >> 5) + (threadIdx.x >> 5); // row r = c*BB + b

  const float4* row = (const float4*)(wf + (size_t)wave * DD);
  float4 p0 = row[lane];        // bytes [lane*16, lane*16+16) of first half
  float4 p1 = row[lane + 32];   // second half of the 1KB row

  float m = fmaxf(fmaxf(fmaxf(p0.x, p0.y), fmaxf(p0.z, p0.w)),
                  fmaxf(fmaxf(p1.x, p1.y), fmaxf(p1.z, p1.w)));
#pragma unroll
  for (int off = 16; off > 0; off >>= 1)
    m = fmaxf(m, __shfl_xor(m, off, 32));

  if (lane == 0) {
    const int c = wave >> 8;         // / BB
    const int b = wave & (BB - 1);   // % BB
    maxsim[b * CC + c] = m;
  }
}

// ---------------------------------------------------------------------------
// Per-(b,c) contribution, branchless so EXEC stays all-1s for WMMA.
//   BCE(x,y) = softplus(x) - x*y   (stable: max(x,0) + log1p(exp(-|x|)))
//   rej      = (1-y) * max(sigmoid(maxsim) - 0.3, 0)
// ---------------------------------------------------------------------------
__device__ __forceinline__ float contrib_f(const float* __restrict__ logits,
                                           const int* __restrict__ labels,
                                           const float* __restrict__ maxsim,
                                           int b, int c) {
  const int idx = b * CC + c;
  const float x  = logits[idx];
  const float y  = (float)labels[idx];
  const float ms = maxsim[idx];
  const float sp  = fmaxf(x, 0.0f) + log1pf(expf(-fabsf(x)));
  const float sig = 1.0f / (1.0f + expf(-ms));
  return sp - x * y + (1.0f - y) * fmaxf(sig - 0.3f, 0.0f);
}

// ---------------------------------------------------------------------------
// Kernel 2: out[b] = sum_c contrib[b,c] via V_WMMA_F32_16X16X4_F32.
// One wave per 16-row b-tile. A(16x4) built on the fly per ISA layout:
//   lane = kh*16 + m  holds A[m, 2*kh] in vgpr0, A[m, 2*kh+1] in vgpr1.
// B = all-ones 4x16 -> every column of D is the row-sum; accumulate over
// 125 K-steps (CC=500 divisible by 4). Column N=0 lives in lane 0 (M=0..7)
// and lane 16 (M=8..15) across the 8 accumulator VGPRs.
// ---------------------------------------------------------------------------
__global__ __launch_bounds__(32) void finalize_wmma_kernel(const float* __restrict__ logits,
                                                           const int* __restrict__ labels,
                                                           const float* __restrict__ maxsim,
                                                           float* __restrict__ out) {
  const int lane = threadIdx.x;      // 0..31, all active
  const int m    = lane & 15;        // A-matrix row within the tile
  const int kh   = lane >> 4;        // 0 -> K={0,1}, 1 -> K={2,3}
  const int b    = blockIdx.x * 16 + m;

  v8f acc = {0.f, 0.f, 0.f, 0.f, 0.f, 0.f, 0.f, 0.f};
  v2f bones;
  bones.x = 1.0f;
  bones.y = 1.0f;

  for (int k0 = 0; k0 < CC; k0 += 4) {
    const int c0 = k0 + kh * 2;
    v2f a;
    a.x = contrib_f(logits, labels, maxsim, b, c0);
    a.y = contrib_f(logits, labels, maxsim, b, c0 + 1);
    // 8 args: (neg_a, A, neg_b, B, c_mod, C, reuse_a, reuse_b)
    acc = __builtin_amdgcn_wmma_f32_16x16x4_f32(false, a, false, bones,
                                                (short)0, acc, false, false);
  }

  // Extract column N=0 of the 16x16 accumulator.
  if (lane == 0) {
#pragma unroll
    for (int v = 0; v < 8; ++v) out[blockIdx.x * 16 + v] = acc[v];
  } else if (lane == 16) {
#pragma unroll
    for (int v = 0; v < 8; ++v) out[blockIdx.x * 16 + 8 + v] = acc[v];
  }
}

extern "C" void kernel_launch(void* const* d_in, const int* in_sizes, int n_in,
                              void* d_out, int out_size, void* d_ws, size_t ws_size,
                              hipStream_t stream) {
  const float* logits = (const float*)d_in[0];   // [B, C] f32
  const float* wf     = (const float*)d_in[1];   // [C, B, D] f32
  const int*   labels = (const int*)d_in[2];     // [B, C] i32
  float* out    = (float*)d_out;                 // [B] f32
  float* maxsim = (float*)d_ws;                  // [B, C] f32 scratch (512000 B)

  // 128000 rows, 8 waves per 256-thread block -> 16000 blocks exactly.
  rowmax_kernel<<<(CC * BB) / 8, 256, 0, stream>>>(wf, maxsim);
  // 256 b-rows / 16 per wave -> 16 single-wave blocks.
  finalize_wmma_kernel<<<BB / 16, 32, 0, stream>>>(logits, labels, maxsim, out);
}